// EmbeddingModule_85770496901401
// MI455X (gfx1250) — compile-verified
//
#include <hip/hip_runtime.h>
#include <hip/hip_bf16.h>

// Problem constants (match reference)
#define NF 26       // features
#define NB 16384    // batch
#define NV 100000   // vocab per table
#define ND 32       // embed dim
#define NH 13       // SE hidden (F/2)

typedef float v2f __attribute__((ext_vector_type(2)));
typedef float v4f __attribute__((ext_vector_type(4)));
typedef float v8f __attribute__((ext_vector_type(8)));

// ---------------------------------------------------------------------------
// Kernel 1: compute gate[0..25] for the LAST batch row only (reference uses
// gate[-1] for every sample). One wave32. First layer (16x32 padded matvec)
// done with V_WMMA_F32_16X16X4_F32, broadcasting the se-vector chunk into all
// 16 columns of B so every column of D equals W1p @ se (robust to column
// striping). Second layer (26x13) is a trivial 13-MAC scalar loop per lane.
// ---------------------------------------------------------------------------
__global__ __launch_bounds__(32) void se_gate_kernel(
    const int* __restrict__ x, const float* __restrict__ tables,
    const float* __restrict__ w1, const float* __restrict__ w2,
    float* __restrict__ gate_out) {
  __shared__ float se[32];
  __shared__ float hsh[16];

  const int L = threadIdx.x;      // lane 0..31
  const int half = L >> 4;        // 0: lanes 0-15, 1: lanes 16-31
  const int m = L & 15;           // M row within half

  // se[f] = mean over D of table_f[x[f, B-1], :]
  float s = 0.f;
  if (L < NF) {
    const int idx = x[(size_t)L * NB + (NB - 1)];
    const float* row = tables + ((size_t)L * NV + (size_t)idx) * ND;
    #pragma unroll
    for (int d = 0; d < ND; ++d) s += row[d];
    s *= (1.0f / 32.0f);
  }
  se[L] = (L < NF) ? s : 0.0f;
  __syncthreads();

  // h16 = W1p(16x32) @ se(32), via 8 chained 16x16x4 F32 WMMAs.
  // A 16x4 f32 layout: VGPR0 = K = 2*half+0, VGPR1 = K = 2*half+1 (M = lane%16).
  // B 4x16 f32 layout mirrors A: row K = k0+2*half+{0,1}; we put the same
  // se value in every lane of a half -> all 16 output columns identical.
  v8f acc = {};
  #pragma unroll
  for (int k0 = 0; k0 < 32; k0 += 4) {
    const int ka = k0 + 2 * half;
    v2f a, b;
    a.x = (m < NH && (ka + 0) < NF) ? w1[m * NF + ka + 0] : 0.0f;
    a.y = (m < NH && (ka + 1) < NF) ? w1[m * NF + ka + 1] : 0.0f;
    b.x = se[ka + 0];
    b.y = se[ka + 1];
    acc = __builtin_amdgcn_wmma_f32_16x16x4_f32(
        /*neg_a=*/false, a, /*neg_b=*/false, b,
        /*c_mod=*/(short)0, acc, /*reuse_a=*/false, /*reuse_b=*/false);
  }

  // D layout: VGPR v, lanes 0-15 -> M=v, lanes 16-31 -> M=v+8. All columns
  // identical, so lane (m==0) of each half publishes its 8 rows.
  if (m == 0) {
    #pragma unroll
    for (int v = 0; v < 8; ++v) hsh[v + 8 * half] = fmaxf(acc[v], 0.0f);
  }
  __syncthreads();

  // gate[f] = sigmoid( sum_j w2[f,j] * h[j] ), f < 26, j < 13
  if (L < NF) {
    float a2 = 0.f;
    #pragma unroll
    for (int j = 0; j < NH; ++j) a2 += w2[L * NH + j] * hsh[j];
    gate_out[L] = 1.0f / (1.0f + expf(-a2));
  }
}

// ---------------------------------------------------------------------------
// Kernel 2: bandwidth-bound gather + scale + stream-out.
// out[b*832 + f*32 + d] = tables[(f*V + x[f,b])*32 + d] * gate[f]
// float4 granularity: o4 = b*208 + f*8 + d4. 8 consecutive lanes read one
// contiguous 128B table row (row base is 128B aligned); stores are perfectly
// coalesced and non-temporal (output is write-once, keep L2 for the tables).
// ---------------------------------------------------------------------------
#define N4 (26u * 16384u * 8u)   // 3,407,872 float4 elements

__global__ __launch_bounds__(256) void gather_scale_kernel(
    const int* __restrict__ x, const float* __restrict__ tables,
    const float* __restrict__ gate, v4f* __restrict__ out) {
  const unsigned o4 = blockIdx.x * 256u + threadIdx.x;
  if (o4 >= N4) return;

  const unsigned b   = o4 / 208u;            // batch row
  const unsigned rem = o4 - b * 208u;
  const unsigned f   = rem >> 3;             // feature
  const unsigned d4  = rem & 7u;             // float4 chunk within row

  const int idx = x[(size_t)f * NB + b];
  const v4f* row = (const v4f*)(tables + ((size_t)f * NV + (size_t)idx) * ND);
  const v4f v = row[d4];
  const float g = gate[f];
  v4f r;
  r.x = v.x * g; r.y = v.y * g; r.z = v.z * g; r.w = v.w * g;
  __builtin_nontemporal_store(r, &out[o4]);
}

extern "C" void kernel_launch(void* const* d_in, const int* in_sizes, int n_in,
                              void* d_out, int out_size, void* d_ws, size_t ws_size,
                              hipStream_t stream) {
  const int*   x      = (const int*)d_in[0];
  const float* tables = (const float*)d_in[1];
  const float* w1     = (const float*)d_in[2];
  const float* w2     = (const float*)d_in[3];
  float* gate = (float*)d_ws;   // 26 floats of scratch

  se_gate_kernel<<<1, 32, 0, stream>>>(x, tables, w1, w2, gate);
  gather_scale_kernel<<<N4 / 256u, 256, 0, stream>>>(x, tables, gate, (v4f*)d_out);
}